// HyperLRPINN_53944789238444
// MI455X (gfx1250) — compile-verified
//
#include <hip/hip_runtime.h>
#include <hip/hip_bf16.h>
#include <math.h>

typedef float v2f __attribute__((ext_vector_type(2)));
typedef float v8f __attribute__((ext_vector_type(8)));

#define HID   64
#define RANK  4
#define NAB   1036          // hypernet output width
#define GSTR  1044          // g-tile row stride (floats, padded to 65*16+4 cols);
                            // 1044 % 64 = 20 -> m*20 mod 64 distinct for m=0..15 (no bank conflicts)
#define XSTR  65            // padded stride for 16x64 activation tiles

// offsets of the low-rank factors inside a g row (matches reference _unpack)
#define OFF_A1 0            // a1[r][d] = g[2r+d]         (4x2)
#define OFF_B1 8            // b1[h][r] = g[8+4h+r]       (64x4)
#define OFF_A2 264          // a2[r][h] = g[264+64r+h]    (4x64)
#define OFF_B2 520          // b2[h][r] = g[520+4h+r]     (64x4)
#define OFF_A3 776          // a3[r][h] = g[776+64r+h]    (4x64)
#define OFF_B3 1032         // b3[0][r] = g[1032+r]       (1x4)

__global__ __launch_bounds__(64) void hyperlr_fused(
    const float* __restrict__ coords, const float* __restrict__ param,
    const float* __restrict__ W1,  const float* __restrict__ B1,
    const float* __restrict__ W2,  const float* __restrict__ B2,
    const float* __restrict__ W3,  const float* __restrict__ B3,
    const float* __restrict__ Wh1, const float* __restrict__ Bh1,
    const float* __restrict__ Wh2, const float* __restrict__ Bh2,
    float* __restrict__ out)
{
    extern __shared__ float smem[];
    const int lane = threadIdx.x & 31;
    const int wave = threadIdx.x >> 5;            // 0..1, each wave owns a 16-sample tile

    float* g_lds  = smem + wave * (16 * GSTR);                      // 16 x 1044 g tile
    float* x_lds  = smem + 2 * (16 * GSTR) + wave * (16 * XSTR);    // 16 x 64 activations
    float* t2_lds = smem + 2 * (16 * GSTR) + 2 * (16 * XSTR) + wave * (16 * RANK);

    const int m    = lane & 15;                   // sample row in A-fragment layout
    const int half = lane >> 4;                   // 0 / 1
    const int kb   = half * 2;                    // k offset within each K=4 group
    const int base = blockIdx.x * 32 + wave * 16; // first sample of this wave's tile

    // ---------- hypernet hidden state as WMMA A-fragments (16 samples x K=64) ----------
    // A layout (32-bit, 16x4): lane m=L&15, k = 4*s + 2*(L>>4) + vgpr
    const float pv = param[base + m];
    v2f afrag[16];
    #pragma unroll
    for (int s = 0; s < 16; ++s) {
        int k = 4 * s + kb;
        afrag[s].x = tanhf(pv * Wh1[k]     + Bh1[k]);
        afrag[s].y = tanhf(pv * Wh1[k + 1] + Bh1[k + 1]);
    }

    // ---------- g = h @ Wh2^T + Bh2 : 65 column tiles of 16 (last padded, stored unconditionally) ----------
    // B layout (4x16): lane n=L&15, k = 2*(L>>4) + vgpr; clamp padding columns to a valid row
    // (garbage only lands in g_lds cols 1036..1043, which are never read).
    auto loadB = [&](int ct, v2f (&dst)[16]) {
        int col  = ct * 16 + m;
        int ccol = col < NAB ? col : (NAB - 1);
        const float* wp = Wh2 + ccol * HID + kb;
        #pragma unroll
        for (int s = 0; s < 16; ++s) { dst[s].x = wp[4 * s]; dst[s].y = wp[4 * s + 1]; }
    };

    v2f bnext[16];
    loadB(0, bnext);
    for (int ct = 0; ct < 65; ++ct) {
        v2f bcur[16];
        #pragma unroll
        for (int s = 0; s < 16; ++s) bcur[s] = bnext[s];
        if (ct < 64) loadB(ct + 1, bnext);        // prefetch next tile while WMMA chain runs

        v8f acc0 = {}, acc1 = {};                 // two chains -> hide WMMA RAW latency
        #pragma unroll
        for (int s = 0; s < 16; s += 2) {
            acc0 = __builtin_amdgcn_wmma_f32_16x16x4_f32(false, afrag[s],     false, bcur[s],
                                                         (short)0, acc0, false, false);
            acc1 = __builtin_amdgcn_wmma_f32_16x16x4_f32(false, afrag[s + 1], false, bcur[s + 1],
                                                         (short)0, acc1, false, false);
        }
        const v8f acc = acc0 + acc1;

        const int col  = ct * 16 + m;
        const int ccol = col < NAB ? col : (NAB - 1);
        const float bias = Bh2[ccol];             // C layout: row = v + 8*half, col = L&15
        #pragma unroll
        for (int v = 0; v < 8; ++v)
            g_lds[(v + 8 * half) * GSTR + col] = acc[v] + bias;
    }
    __syncthreads();

    // ---------- layer 1: x1 = tanh(coords@W1^T + b1*(a1@coords) + B1), built as A-frags ----------
    const float c0 = coords[(base + m) * 2 + 0];
    const float c1 = coords[(base + m) * 2 + 1];
    float t1[RANK];
    #pragma unroll
    for (int r = 0; r < RANK; ++r)
        t1[r] = c0 * g_lds[m * GSTR + OFF_A1 + 2 * r]
              + c1 * g_lds[m * GSTR + OFF_A1 + 2 * r + 1];

    v2f x1frag[16];
    #pragma unroll
    for (int s = 0; s < 16; ++s) {
        #pragma unroll
        for (int j = 0; j < 2; ++j) {
            const int h = 4 * s + kb + j;
            float pre = W1[2 * h] * c0 + W1[2 * h + 1] * c1 + B1[h];
            #pragma unroll
            for (int r = 0; r < RANK; ++r)
                pre += g_lds[m * GSTR + OFF_B1 + 4 * h + r] * t1[r];
            const float x = tanhf(pre);
            x1frag[s][j] = x;
            x_lds[m * XSTR + h] = x;
        }
    }
    __syncthreads();

    // ---------- t2[r] = sum_h x1[h] * a2[r][h] (2 lanes per sample, redundant) ----------
    {
        float t2[RANK] = {0.f, 0.f, 0.f, 0.f};
        for (int h = 0; h < HID; ++h) {
            const float xv = x_lds[m * XSTR + h];
            #pragma unroll
            for (int r = 0; r < RANK; ++r)
                t2[r] += xv * g_lds[m * GSTR + OFF_A2 + 64 * r + h];
        }
        if (half == 0) {
            #pragma unroll
            for (int r = 0; r < RANK; ++r) t2_lds[m * RANK + r] = t2[r];
        }
    }
    __syncthreads();

    // ---------- layer 2: x2 = tanh(x1@W2^T + b2*t2 + B2) via WMMA ----------
    #pragma unroll
    for (int ct = 0; ct < 4; ++ct) {
        const int hcol = ct * 16 + m;             // hidden unit this lane owns in B/C
        const float* wp = W2 + hcol * HID + kb;
        v2f bfrag[16];
        #pragma unroll
        for (int s = 0; s < 16; ++s) { bfrag[s].x = wp[4 * s]; bfrag[s].y = wp[4 * s + 1]; }

        v8f acc0 = {}, acc1 = {};
        #pragma unroll
        for (int s = 0; s < 16; s += 2) {
            acc0 = __builtin_amdgcn_wmma_f32_16x16x4_f32(false, x1frag[s],     false, bfrag[s],
                                                         (short)0, acc0, false, false);
            acc1 = __builtin_amdgcn_wmma_f32_16x16x4_f32(false, x1frag[s + 1], false, bfrag[s + 1],
                                                         (short)0, acc1, false, false);
        }
        const v8f acc = acc0 + acc1;

        const float b2h = B2[hcol];
        #pragma unroll
        for (int v = 0; v < 8; ++v) {
            const int row = v + 8 * half;         // sample index in tile
            float pre = acc[v] + b2h;
            #pragma unroll
            for (int r = 0; r < RANK; ++r)
                pre += t2_lds[row * RANK + r] * g_lds[row * GSTR + OFF_B2 + 4 * hcol + r];
            x_lds[row * XSTR + hcol] = tanhf(pre);
        }
    }
    __syncthreads();

    // ---------- layer 3: out = x2@W3^T + b3*(a3@x2) + B3, one sample per low lane ----------
    if (half == 0) {
        float t3[RANK] = {0.f, 0.f, 0.f, 0.f};
        float w3s = 0.f;
        for (int h = 0; h < HID; ++h) {
            const float xv = x_lds[m * XSTR + h];
            w3s += xv * W3[h];
            #pragma unroll
            for (int r = 0; r < RANK; ++r)
                t3[r] += xv * g_lds[m * GSTR + OFF_A3 + 64 * r + h];
        }
        float o = w3s + B3[0];
        #pragma unroll
        for (int r = 0; r < RANK; ++r)
            o += t3[r] * g_lds[m * GSTR + OFF_B3 + r];
        out[base + m] = o;
    }
}

extern "C" void kernel_launch(void* const* d_in, const int* in_sizes, int n_in,
                              void* d_out, int out_size, void* d_ws, size_t ws_size,
                              hipStream_t stream) {
    const float* coords = (const float*)d_in[0];
    const float* param  = (const float*)d_in[1];
    const float* W1     = (const float*)d_in[2];
    const float* B1     = (const float*)d_in[3];
    const float* W2     = (const float*)d_in[4];
    const float* B2     = (const float*)d_in[5];
    const float* W3     = (const float*)d_in[6];
    const float* B3     = (const float*)d_in[7];
    const float* Wh1    = (const float*)d_in[8];
    const float* Bh1    = (const float*)d_in[9];
    const float* Wh2    = (const float*)d_in[10];
    const float* Bh2    = (const float*)d_in[11];
    float* out = (float*)d_out;

    const int n = in_sizes[1];                    // N (param is [N,1])
    const size_t smem = (size_t)(2 * 16 * GSTR + 2 * 16 * XSTR + 2 * 16 * RANK) * sizeof(float);
    (void)hipFuncSetAttribute((const void*)hyperlr_fused,
                              hipFuncAttributeMaxDynamicSharedMemorySize, (int)smem);

    const int blocks = n / 32;                    // 32 samples per 2-wave block
    hyperlr_fused<<<blocks, 64, smem, stream>>>(coords, param, W1, B1, W2, B2, W3, B3,
                                                Wh1, Bh1, Wh2, Bh2, out);
}